// SmoothLoss_15762529976907
// MI455X (gfx1250) — compile-verified
//
#include <hip/hip_runtime.h>
#include <hip/hip_bf16.h>

typedef __attribute__((ext_vector_type(2))) float v2f;
typedef __attribute__((ext_vector_type(8))) float v8f;

#define NB        8
#define NPTS      4096
#define KKNN      16
#define KBQ       32
#define RK2       0.0625f   /* 0.25^2 */
#define RB2       0.25f     /* 0.5^2  */
#define WKNN      3.0f
#define WBQ       1.0f
#define THREADS   128
#define ROWS_PER_BLOCK 128  /* 4 waves x 32 rows */

__global__ void SmoothLoss_zero_kernel(float* out) { out[0] = 0.0f; }

__global__ __launch_bounds__(THREADS)
void SmoothLoss_15762529976907_kernel(const float* __restrict__ pc,
                                      const float* __restrict__ flow,
                                      float* __restrict__ out)
{
    __shared__ float s_pc[NPTS * 3];          // 48 KB: this batch's points
    __shared__ float s_tile[4][32][17];       // per-wave 32x16 distance tile (padded)

    const int tid  = threadIdx.x;
    const int lane = tid & 31;
    const int wv   = tid >> 5;
    const int b       = blockIdx.x >> 5;                 // 32 blocks per batch
    const int m_block = (blockIdx.x & 31) * ROWS_PER_BLOCK;
    const int m_base  = m_block + wv * 32;               // this wave's 32 rows

    const float* pcb   = pc   + (size_t)b * NPTS * 3;
    const float* flowb = flow + (size_t)b * NPTS * 3;

    for (int i = tid; i < NPTS * 3; i += THREADS) s_pc[i] = pcb[i];
    __syncthreads();

    const int  half = (lane >= 16) ? 1 : 0;
    const int  l15  = lane & 15;

    // ---- A fragments (16x4 f32, K = {x,y,z,0}) for the two 16-row tiles ----
    // Lanes 0-15: M=0..15, VGPR0=K0(x), VGPR1=K1(y); lanes 16-31: VGPR0=K2(z), VGPR1=K3(0)
    const int am0 = m_base + l15;
    const int am1 = am0 + 16;
    v2f a0, a1;
    {
        float x0 = s_pc[3*am0], y0 = s_pc[3*am0+1], z0 = s_pc[3*am0+2];
        float x1 = s_pc[3*am1], y1 = s_pc[3*am1+1], z1 = s_pc[3*am1+2];
        a0.x = half ? z0 : x0;  a0.y = half ? 0.0f : y0;
        a1.x = half ? z1 : x1;  a1.y = half ? 0.0f : y1;
    }

    // ---- row squared norms needed per (VGPR v, half) of the D tiles ----
    float rs0[8], rs1[8];
    {
        const int roff = half * 8;
        #pragma unroll
        for (int v = 0; v < 8; ++v) {
            int m0 = m_base + roff + v;
            float x = s_pc[3*m0], y = s_pc[3*m0+1], z = s_pc[3*m0+2];
            rs0[v] = x*x + y*y + z*z;
            int m1 = m0 + 16;
            x = s_pc[3*m1]; y = s_pc[3*m1+1]; z = s_pc[3*m1+2];
            rs1[v] = x*x + y*y + z*z;
        }
    }

    // ---- per-lane row state: lane owns global row m = m_base + lane ----
    const int   myrow = m_base + lane;
    const float fmx = flowb[3*myrow], fmy = flowb[3*myrow+1], fmz = flowb[3*myrow+2];

    float knn_d[KKNN];
    int   knn_i[KKNN];
    #pragma unroll
    for (int j = 0; j < KKNN; ++j) { knn_d[j] = 1e30f; knn_i[j] = 0; }
    float curmax = 1e30f;

    float bsum = 0.0f, bfirst = 0.0f;
    int   bcnt = 0;

    // ================= main loop over 16-wide column tiles =================
    for (int n0 = 0; n0 < NPTS; n0 += 16) {
        // B fragment (4x16 f32): column n = n0 + (lane&15)
        const int nn = n0 + l15;
        float bx = s_pc[3*nn], by = s_pc[3*nn+1], bz = s_pc[3*nn+2];
        float cs = bx*bx + by*by + bz*bz;
        v2f bf;
        bf.x = half ? bz : bx;
        bf.y = half ? 0.0f : by;

        v8f c0 = {};
        v8f c1 = {};
        c0 = __builtin_amdgcn_wmma_f32_16x16x4_f32(false, a0, false, bf,
                                                   (short)0, c0, false, false);
        c1 = __builtin_amdgcn_wmma_f32_16x16x4_f32(false, a1, false, bf,
                                                   (short)0, c1, false, false);

        // stage clamped squared distances: row = v + 8*half (+16 for tile1), col = lane&15
        const int roff = half * 8;
        #pragma unroll
        for (int v = 0; v < 8; ++v) {
            float d0 = fmaxf(rs0[v] + cs - 2.0f * c0[v], 0.0f);
            float d1 = fmaxf(rs1[v] + cs - 2.0f * c1[v], 0.0f);
            s_tile[wv][roff + v][l15]      = d0;
            s_tile[wv][16 + roff + v][l15] = d1;
        }
        asm volatile("s_wait_dscnt 0" ::: "memory");   // cross-lane RAW within wave

        // ---- each lane consumes its own row's 16 candidates (ascending n) ----
        for (int j = 0; j < 16; ++j) {
            const int   n = n0 + j;
            const float d = s_tile[wv][lane][j];

            // ball query: first KBQ hits in index order (self always qualifies)
            if (bcnt < KBQ && d < RB2) {
                float gx = flowb[3*n], gy = flowb[3*n+1], gz = flowb[3*n+2];
                float l1 = fabsf(fmx - gx) + fabsf(fmy - gy) + fabsf(fmz - gz);
                if (bcnt == 0) bfirst = l1;
                bsum += l1;
                ++bcnt;
            }

            // streaming top-16 smallest (registers; replace-max + rescan)
            if (d < curmax) {
                bool done = false;
                #pragma unroll
                for (int s = 0; s < KKNN; ++s) {
                    if (!done && knn_d[s] == curmax) {
                        knn_d[s] = d; knn_i[s] = n; done = true;
                    }
                }
                float m2 = knn_d[0];
                #pragma unroll
                for (int s = 1; s < KKNN; ++s) m2 = fmaxf(m2, knn_d[s]);
                curmax = m2;
            }
        }
        asm volatile("s_wait_dscnt 0" ::: "memory");   // WAR before next staging
    }

    // ---- finalize KNN term: nearest (= idx[...,0]) + radius fallback ----
    float mind = knn_d[0]; int mini = knn_i[0];
    #pragma unroll
    for (int j = 1; j < KKNN; ++j) {
        bool better = (knn_d[j] < mind) || (knn_d[j] == mind && knn_i[j] < mini);
        mind = better ? knn_d[j] : mind;
        mini = better ? knn_i[j] : mini;
    }
    float f0x = flowb[3*mini], f0y = flowb[3*mini+1], f0z = flowb[3*mini+2];
    float l10 = fabsf(fmx - f0x) + fabsf(fmy - f0y) + fabsf(fmz - f0z);

    float ksum = 0.0f;
    #pragma unroll
    for (int j = 0; j < KKNN; ++j) {
        int nj = knn_i[j];
        float gx = flowb[3*nj], gy = flowb[3*nj+1], gz = flowb[3*nj+2];
        float l1 = fabsf(fmx - gx) + fabsf(fmy - gy) + fabsf(fmz - gz);
        ksum += (knn_d[j] > RK2) ? l10 : l1;   // dist > radius -> use nearest
    }

    // ball-query fill: invalid slots replaced by first hit
    float bq = bsum + ((bcnt < KBQ) ? (float)(KBQ - bcnt) * bfirst : 0.0f);

    float contrib = (WKNN / ((float)NB * NPTS * KKNN)) * ksum
                  + (WBQ  / ((float)NB * NPTS * KBQ )) * bq;

    // wave32 reduction, one atomic per wave
    #pragma unroll
    for (int off = 16; off > 0; off >>= 1)
        contrib += __shfl_down(contrib, off, 32);
    if (lane == 0) atomicAdd(out, contrib);
}

extern "C" void kernel_launch(void* const* d_in, const int* in_sizes, int n_in,
                              void* d_out, int out_size, void* d_ws, size_t ws_size,
                              hipStream_t stream) {
    (void)in_sizes; (void)n_in; (void)out_size; (void)d_ws; (void)ws_size;
    const float* pc   = (const float*)d_in[0];
    const float* flow = (const float*)d_in[1];
    float* out = (float*)d_out;

    SmoothLoss_zero_kernel<<<1, 1, 0, stream>>>(out);
    SmoothLoss_15762529976907_kernel<<<dim3(NB * (NPTS / ROWS_PER_BLOCK)),
                                       dim3(THREADS), 0, stream>>>(pc, flow, out);
}